// Gate_Net_55078660604528
// MI455X (gfx1250) — compile-verified
//
#include <hip/hip_runtime.h>

// -------------------------------------------------------------------------
// Gate_Net: docs = score[score_idx]; per doc build 128x127 sigmoid-gate
// matrices (fwd/bwd) and cumprod along k.  Implemented as
//   cumprod_k(sigmoid(x)) = exp( cumsum_k( logsigmoid(x) ) )
// with the row-wise cumsum done by V_WMMA_F32_16X16X4_F32 against a
// triangular ones B-matrix, carry-propagated across 16-wide N-tiles.
// Memory-bound: 266 MB of f32 output -> ~11.4 us floor at 23.3 TB/s.
// -------------------------------------------------------------------------

typedef __attribute__((ext_vector_type(2))) float v2f;
typedef __attribute__((ext_vector_type(8))) float v8f;

#define N_DOCS   2048
#define DOC_LEN  129
#define LROW     128   // length of f / b
#define KOUT     127   // cumprod output length per row

// log(sigmoid(x)) = min(x,0) - log1p(exp(-|x|)); stable for |x| ~ 1000.
__device__ __forceinline__ float log_sigmoid(float x) {
    float e = __expf(-fabsf(x));
    return fminf(x, 0.0f) - __logf(1.0f + e);
}

__global__ __launch_bounds__(256)
void gate_cumprod_wmma(const float* __restrict__ score,
                       const int*   __restrict__ score_idx,
                       float*       __restrict__ out)
{
    __shared__ float s[DOC_LEN];
    const int doc = blockIdx.x;
    const int dir = blockIdx.y;              // 0 = fwd, 1 = bwd
    const int tid = threadIdx.x;

    if (tid < DOC_LEN) {
        s[tid] = score[score_idx[doc * DOC_LEN + tid]];
    }
    __syncthreads();

    const int lane  = tid & 31;
    const int m     = lane & 15;             // M (A) / N (B,D) sub-index
    const int khalf = lane >> 4;             // K-pair select within a chunk
    const int wave  = tid >> 5;              // 0..7 -> 16-row block
    const int jA    = wave * 16 + m;         // this lane's A-matrix row
    const int jD    = wave * 16 + 8 * khalf; // D-row base for component r

    // reference value: f[j] = s[j] (fwd) or b[j] = s[j+1] (bwd)
    const float ref = s[jA + dir];

    const size_t rowbase =
        ((size_t)dir * N_DOCS + doc) * (size_t)(LROW * KOUT);

    // per-row running cumsum carry, replicated across all lanes
    float carry[8];
#pragma unroll
    for (int r = 0; r < 8; ++r) carry[r] = 0.0f;

    const int bsrc = khalf * 16 + 15;        // lane holding column N=15 of D

#pragma unroll
    for (int n = 0; n < 8; ++n) {            // eight 16-wide N-tiles of k
        // ---- A fragments: 4 diagonal K-chunks (k' = 16n .. 16n+15) ----
        // A 16x4 f32 layout: lanes 0-15 = M; VGPR v holds K = v + 2*khalf
        v2f a[4];
#pragma unroll
        for (int kc = 0; kc < 4; ++kc) {
#pragma unroll
            for (int v = 0; v < 2; ++v) {
                const int kp = n * 16 + kc * 4 + 2 * khalf + v; // global k'
                float hat;
                if (dir == 0) hat = (jA - kp >= 0)       ? s[jA - kp]     : 0.0f;
                else          hat = (jA + kp + 1 < LROW) ? s[jA + kp + 2] : 0.0f;
                a[kc][v] = log_sigmoid((hat - ref) * 100.0f + 5.0f);
            }
        }

        // ---- accumulator starts at the inter-tile carry ----
        v8f d;
#pragma unroll
        for (int r = 0; r < 8; ++r) d[r] = carry[r];

        // ---- intra-tile inclusive cumsum: 4 WMMAs w/ triangular B ----
        // B 4x16 f32 layout: lanes 0-15 = N; VGPR v holds K = v + 2*khalf
#pragma unroll
        for (int kc = 0; kc < 4; ++kc) {
            const int kb = kc * 4 + 2 * khalf;      // k' local to tile
            v2f b;
            b[0] = (kb + 0 <= m) ? 1.0f : 0.0f;     // ones below diagonal
            b[1] = (kb + 1 <= m) ? 1.0f : 0.0f;
            d = __builtin_amdgcn_wmma_f32_16x16x4_f32(
                    false, a[kc], false, b, (short)0, d, false, false);
        }

        // ---- carry out (column N=15) + exp + store ----
        const int kk = n * 16 + m;                  // output k for this lane
#pragma unroll
        for (int r = 0; r < 8; ++r) {
            carry[r] = __shfl(d[r], bsrc, 32);
            if (kk < KOUT) {
                out[rowbase + (size_t)(jD + r) * KOUT + kk] = __expf(d[r]);
            }
        }
    }
}

extern "C" void kernel_launch(void* const* d_in, const int* in_sizes, int n_in,
                              void* d_out, int out_size, void* d_ws, size_t ws_size,
                              hipStream_t stream) {
    const float* score     = (const float*)d_in[0];
    const int*   score_idx = (const int*)  d_in[1];
    float*       out       = (float*)      d_out;

    dim3 grid(N_DOCS, 2);   // (doc, direction)
    dim3 block(256);        // 8 waves of 32; one 16-row block per wave
    gate_cumprod_wmma<<<grid, block, 0, stream>>>(score, score_idx, out);
}